// DyGraph_82154134438715
// MI455X (gfx1250) — compile-verified
//
#include <hip/hip_runtime.h>
#include <hip/hip_bf16.h>
#include <stdint.h>

#define D      20
#define SEQ    8
#define NC     64
#define KC     512
#define ESTR   22          // LDS row stride (floats): 88B, 8B-aligned for b64 async
#define TOPK   10
#define RHO_1  0.015f
#define RHO_2  0.005f

typedef float v2f __attribute__((ext_vector_type(2)));
typedef float v8f __attribute__((ext_vector_type(8)));

// Low 32 bits of a generic pointer to __shared__ == LDS byte offset.
__device__ __forceinline__ uint32_t lds_off(const void* p) {
  return (uint32_t)(uintptr_t)p;
}
// CDNA5 async gather: per-lane global -> LDS, tracked by ASYNCcnt.
__device__ __forceinline__ void async_ld_b64(uint32_t lds_byte, const void* gaddr) {
  asm volatile("global_load_async_to_lds_b64 %0, %1, off"
               :: "v"(lds_byte), "v"((uint64_t)(uintptr_t)gaddr) : "memory");
}
__device__ __forceinline__ void wait_async0() {
  asm volatile("s_wait_asynccnt 0" ::: "memory");
}

// ---------------- Kernel 1: vec_output = mlp2([vecs_use, t0]) over all L locations + norms
__global__ void k_vec_output(const float* __restrict__ vecs, const float* __restrict__ temb,
                             const float* __restrict__ w1, const float* __restrict__ b1,
                             const float* __restrict__ w2, const float* __restrict__ b2,
                             float* __restrict__ vout, float* __restrict__ en,
                             float* __restrict__ tn, int L) {
  int l = blockIdx.x * blockDim.x + threadIdx.x;
  if (l >= L) return;
  float in[2 * D];
  float s_en = 0.f;
  for (int i = 0; i < D; i++) { float v = vecs[(size_t)l * D + i]; in[i] = v; s_en += v * v; }
  for (int i = 0; i < D; i++) in[D + i] = temb[i];   // time_embeddings[0]
  en[l] = s_en;
  float hid[2 * D];
  for (int j = 0; j < 2 * D; j++) hid[j] = b1[j];
  for (int i = 0; i < 2 * D; i++) {
    float v = in[i];
    for (int j = 0; j < 2 * D; j++) hid[j] += v * w1[i * 2 * D + j];
  }
  for (int j = 0; j < 2 * D; j++) hid[j] = fmaxf(hid[j], 0.f);
  float s_tn = 0.f;
  for (int dd = 0; dd < D; dd++) {
    float acc = b2[dd];
    for (int j = 0; j < 2 * D; j++) acc += hid[j] * w2[j * D + dd];
    vout[(size_t)l * D + dd] = acc; s_tn += acc * acc;
  }
  tn[l] = s_tn;
}

// ---------------- Kernel 2: per-token h (history MLP), xi (time MLP), norms, centroid, histogram
__global__ void k_token(const int* __restrict__ x, const int* __restrict__ xts,
                        const int* __restrict__ cassign,
                        const float* __restrict__ vecs, const float* __restrict__ temb,
                        const float* __restrict__ ws1, const float* __restrict__ bs1,
                        const float* __restrict__ ws2, const float* __restrict__ bs2,
                        const float* __restrict__ wo1, const float* __restrict__ bo1,
                        const float* __restrict__ wo2, const float* __restrict__ bo2,
                        float* __restrict__ hfe, float* __restrict__ xif,
                        float* __restrict__ hn, float* __restrict__ xin,
                        int* __restrict__ cent, int* __restrict__ cnt,
                        int N, int user_len) {
  int i = blockIdx.x * blockDim.x + threadIdx.x;
  if (i >= N) return;
  int s = i / user_len, u = i - s * user_len;
  // hist = [shift4, shift3, shift2, shift1, id] -> 100 dims, fed through on the fly
  float hid[D];
  for (int j = 0; j < D; j++) hid[j] = bs1[j];
  for (int seg = 0; seg < 5; seg++) {
    int k = 4 - seg;
    int p = (seg == 4) ? s : (s < k ? s : s - k);
    int idx = x[p * user_len + u];
    const float* e = vecs + (size_t)idx * D;
    for (int dd = 0; dd < D; dd++) {
      float v = e[dd];
      const float* wrow = ws1 + (seg * D + dd) * D;
      for (int j = 0; j < D; j++) hid[j] += v * wrow[j];
    }
  }
  for (int j = 0; j < D; j++) hid[j] = fmaxf(hid[j], 0.f);
  float s_hn = 0.f;
  for (int dd = 0; dd < D; dd++) {
    float acc = bs2[dd];
    for (int j = 0; j < D; j++) acc += hid[j] * ws2[j * D + dd];
    hfe[(size_t)i * D + dd] = acc; s_hn += acc * acc;
  }
  hn[i] = s_hn;
  // xi_out = mlp2([vecs_use[xv], time_embeddings[slot]])
  int xv = x[i];
  float in2[2 * D];
  for (int dd = 0; dd < D; dd++) in2[dd] = vecs[(size_t)xv * D + dd];
  const float* te = temb + (size_t)xts[i] * D;
  for (int dd = 0; dd < D; dd++) in2[D + dd] = te[dd];
  float hid2[2 * D];
  for (int j = 0; j < 2 * D; j++) hid2[j] = bo1[j];
  for (int ii = 0; ii < 2 * D; ii++) {
    float v = in2[ii];
    for (int j = 0; j < 2 * D; j++) hid2[j] += v * wo1[ii * 2 * D + j];
  }
  for (int j = 0; j < 2 * D; j++) hid2[j] = fmaxf(hid2[j], 0.f);
  float s_xn = 0.f;
  for (int dd = 0; dd < D; dd++) {
    float acc = bo2[dd];
    for (int j = 0; j < 2 * D; j++) acc += hid2[j] * wo2[j * D + dd];
    xif[(size_t)i * D + dd] = acc; s_xn += acc * acc;
  }
  xin[i] = s_xn;
  int c = cassign[xv];
  cent[i] = c;
  atomicAdd(&cnt[c], 1);
}

// ---------------- Kernel 3: exclusive scan of 64 bucket counts
__global__ void k_scan(const int* __restrict__ cnt, int* __restrict__ base,
                       int* __restrict__ cursor) {
  if (threadIdx.x == 0) {
    int run = 0;
    for (int c = 0; c < NC; c++) { base[c] = run; cursor[c] = run; run += cnt[c]; }
  }
}

// ---------------- Kernel 4: scatter token ids into centroid buckets
__global__ void k_scatter(const int* __restrict__ cent, int* __restrict__ cursor,
                          int* __restrict__ bucket, int N) {
  int i = blockIdx.x * blockDim.x + threadIdx.x;
  if (i >= N) return;
  int pos = atomicAdd(&cursor[cent[i]], 1);
  bucket[pos] = i;
}

// ---------------- Kernel 5: per-centroid WMMA distance GEMM + top-k + softmax blend
__global__ void __launch_bounds__(256)
k_scores(const int* __restrict__ cands, const int* __restrict__ bucket,
         const int* __restrict__ cnt, const int* __restrict__ basep,
         const float* __restrict__ vecs, const float* __restrict__ vout,
         const float* __restrict__ en_g, const float* __restrict__ tn_g,
         const float* __restrict__ hfe, const float* __restrict__ xif,
         const float* __restrict__ hn_g, const float* __restrict__ xin_g,
         const int* __restrict__ xflat, const float* __restrict__ net,
         float* __restrict__ out) {
  __shared__ float E[KC * ESTR];        // vecs_use[cand]    45056 B
  __shared__ float T[KC * ESTR];        // vec_output[cand]  45056 B
  __shared__ float en_s[KC], tn_s[KC];  // candidate norms
  __shared__ int   cid[KC];             // global candidate ids
  __shared__ float hrow[16 * D], xrow[16 * D];
  __shared__ float hn_s[16], xn_s[16];
  __shared__ int   tok_s[16], xv_s[16];
  __shared__ float sc[16 * KC];         // score tile        32768 B

  const int c    = blockIdx.x;
  const int tid  = threadIdx.x;
  const int lane = tid & 31;
  const int wave = tid >> 5;

  // Stage 1: async-gather candidate embedding rows into LDS (per-lane DMA, ASYNCcnt-tracked)
  for (int r = tid; r < KC; r += blockDim.x) {
    int g = cands[c * KC + r];
    cid[r]  = g;
    en_s[r] = en_g[g];
    tn_s[r] = tn_g[g];
    const float* esrc = vecs + (size_t)g * D;
    const float* tsrc = vout + (size_t)g * D;
    uint32_t eb = lds_off(&E[r * ESTR]);
    uint32_t tb = lds_off(&T[r * ESTR]);
    #pragma unroll
    for (int q = 0; q < D / 2; q++) {         // 10 x b64 per 80-byte row
      async_ld_b64(eb + q * 8, esrc + q * 2);
      async_ld_b64(tb + q * 8, tsrc + q * 2);
    }
  }
  wait_async0();
  __syncthreads();

  const int nt    = cnt[c];
  const int cb    = basep[c];
  const int tiles = (nt + 15) >> 4;
  const int m16   = lane & 15;
  const int hi    = lane >> 4;    // K-half selector in 32-bit WMMA fragments

  for (int tile = 0; tile < tiles; ++tile) {
    if (tid < 16) {
      int slot = tile * 16 + tid;
      if (slot < nt) {
        int tk = bucket[cb + slot];
        tok_s[tid] = tk; xv_s[tid] = xflat[tk];
        hn_s[tid] = hn_g[tk]; xn_s[tid] = xin_g[tk];
        for (int dd = 0; dd < D; dd++) {
          hrow[tid * D + dd] = hfe[(size_t)tk * D + dd];
          xrow[tid * D + dd] = xif[(size_t)tk * D + dd];
        }
      } else {
        tok_s[tid] = -1; xv_s[tid] = 0; hn_s[tid] = 0.f; xn_s[tid] = 0.f;
        for (int dd = 0; dd < D; dd++) { hrow[tid * D + dd] = 0.f; xrow[tid * D + dd] = 0.f; }
      }
    }
    __syncthreads();

    // Stage 2: 8 waves x 4 candidate tiles; 16x16 WMMA dot products, K=20 in 5 chunks of 4
    for (int ci = 0; ci < 4; ++ci) {
      int ct = wave * 4 + ci;
      v8f acc_e = {0.f,0.f,0.f,0.f,0.f,0.f,0.f,0.f};
      v8f acc_t = {0.f,0.f,0.f,0.f,0.f,0.f,0.f,0.f};
      #pragma unroll
      for (int k0 = 0; k0 < D; k0 += 4) {
        int kk = k0 + hi * 2;
        v2f a, b;
        a.x = hrow[m16 * D + kk];           a.y = hrow[m16 * D + kk + 1];
        b.x = E[(ct * 16 + m16) * ESTR + kk]; b.y = E[(ct * 16 + m16) * ESTR + kk + 1];
        acc_e = __builtin_amdgcn_wmma_f32_16x16x4_f32(false, a, false, b,
                                                      (short)0, acc_e, false, false);
        a.x = xrow[m16 * D + kk];           a.y = xrow[m16 * D + kk + 1];
        b.x = T[(ct * 16 + m16) * ESTR + kk]; b.y = T[(ct * 16 + m16) * ESTR + kk + 1];
        acc_t = __builtin_amdgcn_wmma_f32_16x16x4_f32(false, a, false, b,
                                                      (short)0, acc_t, false, false);
      }
      int n = ct * 16 + m16;
      float en_v = en_s[n], tn_v = tn_s[n];
      #pragma unroll
      for (int vg = 0; vg < 8; ++vg) {
        int m = vg + (hi << 3);
        float de = sqrtf(fmaxf(hn_s[m] + en_v - 2.f * acc_e[vg], 0.f));
        float dt = sqrtf(fmaxf(xn_s[m] + tn_v - 2.f * acc_t[vg], 0.f));
        sc[m * KC + n] = __expf(-RHO_1 * de - RHO_2 * dt);
      }
    }
    __syncthreads();

    // Stage 3: iterative wave-parallel top-10 + softmax(+self=1.0) + blended output
    for (int mi = 0; mi < 2; ++mi) {
      int m  = wave * 2 + mi;
      int tk = tok_s[m];
      float vals[TOPK]; int cols[TOPK];
      for (int t = 0; t < TOPK; t++) {
        float bv = -1e30f; int bi = 0x7fffffff;
        #pragma unroll
        for (int j = 0; j < KC / 32; j++) {
          int col = lane + 32 * j;
          float v = sc[m * KC + col];
          if (v > bv || (v == bv && col < bi)) { bv = v; bi = col; }
        }
        #pragma unroll
        for (int off = 16; off >= 1; off >>= 1) {
          float ov = __shfl_xor(bv, off, 32);
          int   oi = __shfl_xor(bi, off, 32);
          if (ov > bv || (ov == bv && oi < bi)) { bv = ov; bi = oi; }
        }
        vals[t] = bv; cols[t] = bi;
        if ((bi & 31) == lane) sc[m * KC + bi] = -1e30f;   // mark used
      }
      if (tk >= 0 && lane < D) {
        float mx  = fmaxf(vals[0], 1.0f);
        float den = __expf(1.0f - mx);
        float w[TOPK];
        #pragma unroll
        for (int t = 0; t < TOPK; t++) { w[t] = __expf(vals[t] - mx); den += w[t]; }
        float acc = (__expf(1.0f - mx) / den) * net[(size_t)xv_s[m] * D + lane];
        #pragma unroll
        for (int t = 0; t < TOPK; t++)
          acc += (w[t] / den) * net[(size_t)cid[cols[t]] * D + lane];
        out[(size_t)tk * D + lane] = acc;
      }
    }
    __syncthreads();
  }
}

// ---------------- Host-side launcher
extern "C" void kernel_launch(void* const* d_in, const int* in_sizes, int n_in,
                              void* d_out, int out_size, void* d_ws, size_t ws_size,
                              hipStream_t stream) {
  const int*   x       = (const int*)d_in[0];
  const int*   xts     = (const int*)d_in[1];
  const int*   cands   = (const int*)d_in[2];
  const int*   cassign = (const int*)d_in[3];
  const float* vecs    = (const float*)d_in[4];
  const float* net     = (const float*)d_in[5];
  const float* temb    = (const float*)d_in[6];
  const float* ws1     = (const float*)d_in[7];
  const float* bs1     = (const float*)d_in[8];
  const float* ws2     = (const float*)d_in[9];
  const float* bs2     = (const float*)d_in[10];
  const float* wo1     = (const float*)d_in[11];
  const float* bo1     = (const float*)d_in[12];
  const float* wo2     = (const float*)d_in[13];
  const float* bo2     = (const float*)d_in[14];
  const float* wi1     = (const float*)d_in[15];
  const float* bi1     = (const float*)d_in[16];
  const float* wi2     = (const float*)d_in[17];
  const float* bi2     = (const float*)d_in[18];
  float* out = (float*)d_out;

  const int N        = in_sizes[0];
  const int user_len = N / SEQ;
  const int L        = in_sizes[4] / D;

  char* w = (char*)d_ws;
  size_t off = 0;
  auto alloc = [&](size_t bytes) -> void* {
    void* p = w + off;
    off = (off + bytes + 255) & ~((size_t)255);
    return p;
  };
  float* vout   = (float*)alloc((size_t)L * D * 4);
  float* en     = (float*)alloc((size_t)L * 4);
  float* tn     = (float*)alloc((size_t)L * 4);
  float* hfe    = (float*)alloc((size_t)N * D * 4);
  float* xif    = (float*)alloc((size_t)N * D * 4);
  float* hn     = (float*)alloc((size_t)N * 4);
  float* xin    = (float*)alloc((size_t)N * 4);
  int*   cent   = (int*)alloc((size_t)N * 4);
  int*   bucket = (int*)alloc((size_t)N * 4);
  int*   cnt    = (int*)alloc(NC * 4);
  int*   base   = (int*)alloc(NC * 4);
  int*   cursor = (int*)alloc(NC * 4);
  (void)ws_size; (void)n_in; (void)out_size;
  (void)bi1; (void)bi2; // consumed below

  hipMemsetAsync(cnt, 0, NC * sizeof(int), stream);

  k_vec_output<<<(L + 255) / 256, 256, 0, stream>>>(vecs, temb, wi1, bi1, wi2, bi2,
                                                    vout, en, tn, L);
  k_token<<<(N + 255) / 256, 256, 0, stream>>>(x, xts, cassign, vecs, temb,
                                               ws1, bs1, ws2, bs2,
                                               wo1, bo1, wo2, bo2,
                                               hfe, xif, hn, xin, cent, cnt,
                                               N, user_len);
  k_scan<<<1, 32, 0, stream>>>(cnt, base, cursor);
  k_scatter<<<(N + 255) / 256, 256, 0, stream>>>(cent, cursor, bucket, N);
  k_scores<<<NC, 256, 0, stream>>>(cands, bucket, cnt, base,
                                   vecs, vout, en, tn,
                                   hfe, xif, hn, xin,
                                   x, net, out);
}